// StrainEnergyLoss_4355096838220
// MI455X (gfx1250) — compile-verified
//
#include <hip/hip_runtime.h>

// StrainEnergyLoss on MI455X (gfx1250, wave32).
// Bandwidth-bound streaming reduction: 268 MB @ 23.3 TB/s => ~11.5 us floor.
// Fused formulation: S_b = sum_j w_j * (pred[b,j]-true[b,j]),
//   w_j = 0.5*([1<=j<=idx] + [j<idx]);  out = mean_b(S_b^2).

typedef float v2f __attribute__((ext_vector_type(2)));
typedef float v4f __attribute__((ext_vector_type(4)));
typedef float v8f __attribute__((ext_vector_type(8)));

#define BDIM 8192
#define NDIM 4096
#define THREADS 256                    // 8 waves (wave32)
#define WAVES (THREADS / 32)
#define VEC_ITERS (NDIM / (THREADS * 4))   // 4 x float4 per thread

__global__ __launch_bounds__(THREADS)
void row_integral_kernel(const float* __restrict__ y_pred,
                         const float* __restrict__ y_true,
                         const int*   __restrict__ fracture_idx,
                         float* __restrict__ row_sum)
{
    const int b   = blockIdx.x;
    const int tid = threadIdx.x;
    const int idx = fracture_idx[b];                 // per-row scalar (SGPR)
    const float* __restrict__ prow = y_pred + (size_t)b * NDIM;
    const float* __restrict__ trow = y_true + (size_t)b * NDIM;

    // Two per-lane accumulators -> 2 VGPRs = the A operand of a 16x4 f32 WMMA.
    float acc0 = 0.0f, acc1 = 0.0f;

    #pragma unroll
    for (int k = 0; k < VEC_ITERS; ++k) {
        const int j0 = (tid + k * THREADS) * 4;      // coalesced b128 loads
        v4f p = __builtin_nontemporal_load((const v4f*)(prow + j0));
        v4f t = __builtin_nontemporal_load((const v4f*)(trow + j0));
        #pragma unroll
        for (int c = 0; c < 4; ++c) {
            const int j = j0 + c;
            // element j is right end of segment j-1 (if 1<=j<=idx) and
            // left end of segment j (if j<idx); each contributes 0.5
            const float w = 0.5f * ((float)((j >= 1) & (j <= idx)) +
                                    (float)(j < idx));
            const float d = p[c] - t[c];
            if (c & 1) acc1 = fmaf(w, d, acc1);
            else       acc0 = fmaf(w, d, acc0);
        }
    }

    // Cross-lane reduction of the wave's 64 partials in ONE
    // v_wmma_f32_16x16x4_f32:  D = A x ones + 0.
    // With B = all-ones, D[m,n] = sum_k A[m,k] for every column n, so the
    // placement of partials inside A is irrelevant. Summing the 8 D VGPRs in
    // each lane gives that lane's half-column sum; adding lane j to lane j^16
    // (columns are identical) yields the full wave sum.
    v2f a;    a.x = acc0;  a.y = acc1;
    v2f ones; ones.x = 1.0f; ones.y = 1.0f;
    v8f c = {};
    c = __builtin_amdgcn_wmma_f32_16x16x4_f32(
            /*neg_a=*/false, a, /*neg_b=*/false, ones,
            /*c_mod=*/(short)0, c, /*reuse_a=*/false, /*reuse_b=*/false);

    float s = c[0] + c[1] + c[2] + c[3] + c[4] + c[5] + c[6] + c[7];
    const float other = __shfl_xor(s, 16, 32);       // wave32
    const float wave_sum = s + other;

    __shared__ float lds[WAVES];
    const int wave = tid >> 5;
    const int lane = tid & 31;
    if (lane == 0) lds[wave] = wave_sum;
    __syncthreads();
    if (tid == 0) {
        float tot = 0.0f;
        #pragma unroll
        for (int w = 0; w < WAVES; ++w) tot += lds[w];  // fixed order
        row_sum[b] = tot;
    }
}

__global__ __launch_bounds__(1024)
void mse_reduce_kernel(const float* __restrict__ row_sum,
                       float* __restrict__ out)
{
    __shared__ float lds[1024];
    const int tid = threadIdx.x;
    float acc = 0.0f;
    #pragma unroll
    for (int i = 0; i < BDIM / 1024; ++i) {           // fixed-order strides
        const float v = row_sum[tid + i * 1024];
        acc = fmaf(v, v, acc);
    }
    lds[tid] = acc;
    __syncthreads();
    #pragma unroll
    for (int stride = 512; stride > 0; stride >>= 1) { // deterministic tree
        if (tid < stride) lds[tid] += lds[tid + stride];
        __syncthreads();
    }
    if (tid == 0) out[0] = lds[0] * (1.0f / (float)BDIM);
}

extern "C" void kernel_launch(void* const* d_in, const int* in_sizes, int n_in,
                              void* d_out, int out_size, void* d_ws, size_t ws_size,
                              hipStream_t stream) {
    const float* y_pred = (const float*)d_in[0];
    const float* y_true = (const float*)d_in[1];
    const int*   fidx   = (const int*)d_in[2];   // harness passes integer inputs as int32
    float* row_sum = (float*)d_ws;               // BDIM floats of scratch (32 KB)

    row_integral_kernel<<<BDIM, THREADS, 0, stream>>>(y_pred, y_true, fidx, row_sum);
    mse_reduce_kernel<<<1, 1024, 0, stream>>>(row_sum, (float*)d_out);
}